// GeneralizedMeanPooling_38474317037946
// MI455X (gfx1250) — compile-verified
//
#include <hip/hip_runtime.h>
#include <cstdint>
#include <cstddef>

// ---------------------------------------------------------------------------
// GeM pooling, MI455X (gfx1250).
// Memory-bound: 1.02 GB of x at 23.3 TB/s => ~44us floor. Strategy:
//  - sorted `batch` => each block owns a contiguous row chunk, flushes per-
//    segment partial sums with global f32 atomics only at segment boundaries.
//  - wave-private async global->LDS pipeline (ASYNCcnt), 4-deep, zero
//    workgroup barriers in the steady-state loop.
//  - pow via v_log_f32 / v_exp_f32 (TRANS ops co-execute with VALU).
// ---------------------------------------------------------------------------

#define DCOLS   256
#define GSEG    1024
#define EPSV    1e-6f
#define BLOCK   256           // 8 waves (wave32)
#define NWAVE   8
#define CHUNK   1024          // rows per block
#define TILE    16            // rows per async tile (per wave: 16 x 128B = 2KB)
#define NBUF    4             // 4-deep pipeline: 12 b128 async ops in flight/wave

#if defined(__HIP_DEVICE_COMPILE__) && \
    __has_builtin(__builtin_amdgcn_global_load_async_to_lds_b128) && \
    __has_builtin(__builtin_amdgcn_s_wait_asynccnt)
#define USE_ASYNC 1
#else
#define USE_ASYNC 0
#endif

// Builtin signature (probed via toolchain diagnostic): arg0 = v4i* in global
// address space (AS1), arg1 = v4i* in LDS address space (AS3), then two imms.
typedef int v4i __attribute__((ext_vector_type(4)));
typedef __attribute__((address_space(1))) v4i* gv4_t;   // global v4i*
typedef __attribute__((address_space(3))) v4i* lv4_t;   // LDS v4i*

__device__ __forceinline__ float pow_pos(float y, float p) {
    // y > 0 guaranteed (clamped to EPS). pow(y,p) = exp2(p * log2(y)).
    return __builtin_amdgcn_exp2f(p * __builtin_amdgcn_logf(y));
}

__device__ __forceinline__ void async_cp16(const float* gsrc, float* ldst) {
#if USE_ASYNC
    __builtin_amdgcn_global_load_async_to_lds_b128(
        (gv4_t)const_cast<float*>(gsrc), (lv4_t)ldst, /*offset=*/0, /*cpol=*/0);
#else
    *(float4*)ldst = *(const float4*)gsrc;  // synchronous fallback
#endif
}

// ---------------------------------------------------------------------------
// Streaming partial sums.
//   Block b owns rows [b*CHUNK, b*CHUNK + rows). Wave w owns columns
//   [32w, 32w+32); lane owns one column. Tiles of 16 rows are async-copied
//   into wave-private LDS slices, quad buffered. The async prologue is issued
//   BEFORE the segment-id staging so the DMA overlaps it.
// ---------------------------------------------------------------------------
__global__ __launch_bounds__(BLOCK) void gemp_partial(
    const float* __restrict__ x, const int* __restrict__ batch,
    const float* __restrict__ p_ptr, float* __restrict__ sums,
    float* __restrict__ counts, int N)
{
    __shared__ float tilebuf[NBUF][NWAVE][TILE][32];   // 64 KB
    __shared__ int   seg_lds[CHUNK];                   // 4 KB

    const int tid  = threadIdx.x;
    const int wave = tid >> 5;
    const int lane = tid & 31;

    const long long row0 = (long long)blockIdx.x * CHUNK;
    long long remain = (long long)N - row0;
    if (remain <= 0) return;
    const int rows = (remain < (long long)CHUNK) ? (int)remain : CHUNK;

    // Per-wave async staging of tile t into LDS buffer b (4 x b128 per wave).
    const int sub   = lane >> 3;            // row offset within a 4-row group
    const int seg16 = lane & 7;             // which 16B of the 128B row slice
    // Per-lane constant parts of the addresses (computed once).
    const float* gbase = x + (size_t)row0 * DCOLS + wave * 32 + seg16 * 4;

    // Full tiles: every row in-bounds -> no clamp, pure strided addresses.
    auto issue_full = [&](int t, int b) {
        #pragma unroll
        for (int i = 0; i < 4; ++i) {
            const int r_in_tile = i * 4 + sub;
            const float* gsrc = gbase + (size_t)(t * TILE + r_in_tile) * DCOLS;
            async_cp16(gsrc, &tilebuf[b][wave][r_in_tile][seg16 * 4]);
        }
    };
    // Tail tile: clamp row to the last valid row (duplicates are ignored).
    auto issue_tail = [&](int t, int b) {
        #pragma unroll
        for (int i = 0; i < 4; ++i) {
            const int r_in_tile = i * 4 + sub;
            int r = t * TILE + r_in_tile;
            if (r > rows - 1) r = rows - 1;
            async_cp16(gbase + (size_t)r * DCOLS, &tilebuf[b][wave][r_in_tile][seg16 * 4]);
        }
    };
    auto issue_tile = [&](int t, int b) {
        if ((t + 1) * TILE <= rows) issue_full(t, b);   // uniform scalar branch
        else                        issue_tail(t, b);
    };

    // Kick off the DMA pipeline first: these depend only on row0/rows, so the
    // copies fly while we stage segment ids below.
    const int ntiles = (rows + TILE - 1) / TILE;
    issue_tile(0, 0);
    if (ntiles > 1) issue_tile(1, 1);
    if (ntiles > 2) issue_tile(2, 2);

    const float p = p_ptr[0];

    // Stage this chunk's segment ids into LDS (one barrier, then read-only).
    #pragma unroll
    for (int k = 0; k < CHUNK / BLOCK; ++k) {
        const int i = tid + k * BLOCK;
        const long long r = row0 + i;
        seg_lds[i] = (r < (long long)N) ? batch[r] : -1;
    }
    __syncthreads();

    float acc    = 0.0f;
    int   runlen = 0;
    const int col = wave * 32 + lane;
    int seg_next = seg_lds[0];

    for (int t = 0; t < ntiles; ++t) {
        const int b = t & (NBUF - 1);
#if USE_ASYNC
        // Async loads complete in order: waiting until only the (up to) two
        // younger tiles' 4+4 ops remain guarantees tile t has landed.
        if      (t + 2 < ntiles) __builtin_amdgcn_s_wait_asynccnt(8);
        else if (t + 1 < ntiles) __builtin_amdgcn_s_wait_asynccnt(4);
        else                     __builtin_amdgcn_s_wait_asynccnt(0);
#endif
        const int base = t * TILE;
        const int rlim = (rows - base < TILE) ? (rows - base) : TILE;
        #pragma unroll 4
        for (int r = 0; r < rlim; ++r) {
            const float v = tilebuf[b][wave][r][lane];
            acc += pow_pos(fmaxf(v, EPSV), p);
            ++runlen;
            const int idx = base + r;
            const int seg = seg_next;
            const bool last = (idx + 1 >= rows);
            seg_next = last ? -1 : seg_lds[idx + 1];
            if (last || seg_next != seg) {     // uniform branch across block
                atomicAdd(sums + (size_t)seg * DCOLS + col, acc);
                if (tid == 0) atomicAdd(counts + seg, (float)runlen);
                acc = 0.0f;
                runlen = 0;
            }
        }
        // Buffer b was only read by this wave; its DS reads are consumed by
        // the accumulates above, so it is safe to refill it now.
        if (t + 3 < ntiles) issue_tile(t + 3, b);
    }
}

// ---------------------------------------------------------------------------
// Zero the accumulators (harness poisons d_out / d_ws with 0xAA).
// ---------------------------------------------------------------------------
__global__ __launch_bounds__(BLOCK) void gemp_zero(float* __restrict__ sums,
                                                   float* __restrict__ counts) {
    const int i = blockIdx.x * BLOCK + threadIdx.x;
    if (i < GSEG * DCOLS) sums[i] = 0.0f;
    if (i < GSEG)         counts[i] = 0.0f;
}

// ---------------------------------------------------------------------------
// Finalize in place: out = (sum / max(count,1))^(1/p).
// ---------------------------------------------------------------------------
__global__ __launch_bounds__(BLOCK) void gemp_final(float* __restrict__ sums,
                                                    const float* __restrict__ counts,
                                                    const float* __restrict__ p_ptr) {
    const int i = blockIdx.x * BLOCK + threadIdx.x;
    if (i >= GSEG * DCOLS) return;
    const float inv_p = 1.0f / p_ptr[0];
    const float m = sums[i] / fmaxf(counts[i >> 8], 1.0f);  // D == 256
    sums[i] = __builtin_amdgcn_exp2f(inv_p * __builtin_amdgcn_logf(m));
}

// ---------------------------------------------------------------------------
extern "C" void kernel_launch(void* const* d_in, const int* in_sizes, int n_in,
                              void* d_out, int out_size, void* d_ws, size_t ws_size,
                              hipStream_t stream) {
    const float* x     = (const float*)d_in[0];
    const int*   batch = (const int*)d_in[1];
    const float* p     = (const float*)d_in[2];
    const int    N     = in_sizes[1];          // rows (batch has N elements)

    float* sums   = (float*)d_out;             // [GSEG * DCOLS] accumulators
    float* counts = (float*)d_ws;              // [GSEG] float counts (4 KB)

    const int nzero = (GSEG * DCOLS + BLOCK - 1) / BLOCK;
    gemp_zero<<<nzero, BLOCK, 0, stream>>>(sums, counts);

    const int nblocks = (N + CHUNK - 1) / CHUNK;
    gemp_partial<<<nblocks, BLOCK, 0, stream>>>(x, batch, p, sums, counts, N);

    gemp_final<<<nzero, BLOCK, 0, stream>>>(sums, counts, p);
}